// Hippocampus_31722628448627
// MI455X (gfx1250) — compile-verified
//
#include <hip/hip_runtime.h>
#include <hip/hip_bf16.h>

// Problem constants (match reference)
#define Bn 32
#define Tn 256
#define Dn 2048
#define Hn 16
#define DKn 64
#define PROJ_N 1024   // H*DK

typedef __attribute__((ext_vector_type(2))) float v2f;
typedef __attribute__((ext_vector_type(8))) float v8f;
typedef __attribute__((ext_vector_type(4))) int v4i;
typedef __attribute__((ext_vector_type(4))) unsigned int u32x4;
typedef __attribute__((ext_vector_type(4))) int i32x4;
typedef __attribute__((ext_vector_type(8))) int i32x8;

static __device__ __forceinline__ v8f vz8() {
  v8f z = {0.f,0.f,0.f,0.f,0.f,0.f,0.f,0.f};
  return z;
}

// D = A(16x4) * B(4x16) + C, fp32 WMMA
static __device__ __forceinline__ v8f wmma4(v2f a, v2f b, v8f c) {
  return __builtin_amdgcn_wmma_f32_16x16x4_f32(
      false, a, false, b, (short)0, c, false, false);
}

// ---------------------------------------------------------------------------
// CDNA5 async / tensor data-movement wrappers (with fallbacks)
// ---------------------------------------------------------------------------
#if __has_builtin(__builtin_amdgcn_s_wait_asynccnt)
#define WAIT_ASYNC(n) __builtin_amdgcn_s_wait_asynccnt(n)
#else
#define WAIT_ASYNC(n) asm volatile("s_wait_asynccnt " #n ::: "memory")
#endif

#if __has_builtin(__builtin_amdgcn_s_wait_tensorcnt)
#define WAIT_TENSOR0() __builtin_amdgcn_s_wait_tensorcnt(0)
#else
#define WAIT_TENSOR0() asm volatile("s_wait_tensorcnt 0x0" ::: "memory")
#endif

// 16-byte global -> LDS async copy (ASYNCcnt-tracked, no VGPR round trip)
static __device__ __forceinline__ void async_copy16(const float* g, float* l) {
#if __has_builtin(__builtin_amdgcn_global_load_async_to_lds_b128)
  float* gnc = const_cast<float*>(g);
  __builtin_amdgcn_global_load_async_to_lds_b128(
      (__attribute__((address_space(1))) v4i*)gnc,
      (__attribute__((address_space(3))) v4i*)l, 0, 0);
#else
  unsigned lo = (unsigned)(uintptr_t)(__attribute__((address_space(3))) void*)l;
  unsigned long long ga = (unsigned long long)(uintptr_t)g;
  asm volatile("global_load_async_to_lds_b128 %0, %1, off"
               :: "v"(lo), "v"(ga) : "memory");
#endif
}

#if __has_builtin(__builtin_amdgcn_tensor_load_to_lds)
#define HAS_TDM 1
// 2-D tile load via Tensor Data Mover.  Element size fixed at 4 bytes.
// lds_off: LDS byte offset; gaddr: global byte address of tile start;
// tile_w/tile_h in elements; row_stride in elements;
// pad_int_code/pad_amt_code per D# encoding (interval 0=2DW..5=64DW; amount 0=1DW,1=2DW)
static __device__ __forceinline__ void tdm_load_2d(unsigned lds_off,
                                                   unsigned long long gaddr,
                                                   int tile_w, int tile_h,
                                                   unsigned long long row_stride,
                                                   int pad_int_code, int pad_amt_code,
                                                   int pad_en) {
  u32x4 g0;
  g0[0] = 1u;                                            // count=1, user D#
  g0[1] = lds_off;                                       // lds_addr
  g0[2] = (unsigned)(gaddr & 0xffffffffull);             // global_addr[31:0]
  g0[3] = (unsigned)((gaddr >> 32) & 0x01ffffffull)      // global_addr[56:32]
        | 0x80000000u;                                   // type=2 ("image")
  i32x8 g1;
  g1[0] = (2 << 16)                                      // data_size = 4B
        | (pad_en << 20)
        | (pad_int_code << 22)
        | (pad_amt_code << 25);
  g1[1] = (int)((unsigned)(tile_w) << 16);               // tensor_dim0[15:0] (=tile_w)
  g1[2] = (int)((unsigned)(tile_h) << 16);               // dim0 hi=0 | tensor_dim1[15:0]
  g1[3] = (int)((unsigned)(tile_w) << 16);               // dim1 hi=0 | tile_dim0
  g1[4] = tile_h & 0xffff;                               // tile_dim1 | tile_dim2=0
  g1[5] = (int)(row_stride & 0xffffffffull);             // tensor_dim0_stride[31:0]
  g1[6] = (int)((row_stride >> 32) & 0xffffull);         // stride[47:32] | dim1_stride lo = 0
  g1[7] = 0;
  i32x4 z4 = {0, 0, 0, 0};
#if __clang_major__ >= 23
  i32x8 z8 = {0, 0, 0, 0, 0, 0, 0, 0};
  __builtin_amdgcn_tensor_load_to_lds(g0, g1, z4, z4, z8, 0);
#else
  __builtin_amdgcn_tensor_load_to_lds(g0, g1, z4, z4, 0);
#endif
}
static __device__ __forceinline__ unsigned lds_offset_of(const void* p) {
  return (unsigned)(uintptr_t)(__attribute__((address_space(3))) const void*)p;
}
#endif

// ---------------------------------------------------------------------------
// Kernel 1: projection GEMM  Y[M,N] = X[M,K] * W[N,K]^T
// M=8192, N=1024, K=2048.  WG = 256 thr (8 waves). WG tile 32(M) x 256(N).
// Wave tile 16 x 64. K-block 32, double-buffered via async global->LDS copies.
// ---------------------------------------------------------------------------
__global__ __launch_bounds__(256) void proj_gemm(const float* __restrict__ X,
                                                 const float* __restrict__ W,
                                                 float* __restrict__ Y) {
  __shared__ float As[2][32][36];   // pad 4 -> 16B-aligned rows, bank-shifted
  __shared__ float Bs[2][256][36];

  const int m0 = blockIdx.x * 32;
  const int n0 = blockIdx.y * 256;
  const int tid  = threadIdx.x;
  const int lane = tid & 31;
  const int wave = tid >> 5;
  const int mw = (wave >> 2) * 16;
  const int nw = (wave & 3) * 64;
  const int ml = lane & 15;
  const int hi = lane >> 4;
  const int kb = hi * 2;
  const int rowL = tid >> 3, colL = (tid & 7) * 4;   // staging layout

  v8f acc[4];
#pragma unroll
  for (int j = 0; j < 4; j++) acc[j] = vz8();

  // issue one K-block of async copies (9 x 16B per thread)
  auto stage = [&](int k0, int buf) {
    async_copy16(X + (size_t)(m0 + rowL) * Dn + k0 + colL, &As[buf][rowL][colL]);
#pragma unroll
    for (int i = 0; i < 8; i++) {
      const int rr = rowL + i * 32;
      async_copy16(W + (size_t)(n0 + rr) * Dn + k0 + colL, &Bs[buf][rr][colL]);
    }
  };

  stage(0, 0);
  int ping = 0;
  for (int k0 = 0; k0 < Dn; k0 += 32) {
    if (k0 + 32 < Dn) {
      stage(k0 + 32, ping ^ 1);
      WAIT_ASYNC(9);          // previous block's 9 copies done; next 9 in flight
    } else {
      WAIT_ASYNC(0);
    }
    __syncthreads();
#pragma unroll
    for (int kk = 0; kk < 32; kk += 4) {
      const v2f a = *(const v2f*)&As[ping][mw + ml][kk + kb];
#pragma unroll
      for (int j = 0; j < 4; j++) {
        const v2f b = *(const v2f*)&Bs[ping][nw + j * 16 + ml][kk + kb];
        acc[j] = wmma4(a, b, acc[j]);
      }
    }
    __syncthreads();          // before this buffer is refilled next round
    ping ^= 1;
  }
  // store D per C-layout: VGPR r -> (M = r + hi*8, N = lane&15)
#pragma unroll
  for (int j = 0; j < 4; j++)
#pragma unroll
    for (int r = 0; r < 8; r++) {
      const int row = m0 + mw + r + hi * 8;
      const int col = n0 + nw + j * 16 + ml;
      Y[(size_t)row * PROJ_N + col] = acc[j][r];
    }
}

// ---------------------------------------------------------------------------
// Kernel 2: beta/gamma logits.  One block per (b,t) row.
// ---------------------------------------------------------------------------
__global__ __launch_bounds__(256) void proj_small(const float* __restrict__ content,
                                                  const float* __restrict__ b_w,
                                                  const float* __restrict__ b_b,
                                                  const float* __restrict__ ts_w,
                                                  const float* __restrict__ ts_b,
                                                  const float* __restrict__ hd,
                                                  float* __restrict__ bl,
                                                  float* __restrict__ gl) {
  __shared__ float row[Dn];
  __shared__ float part[32][9];
  const int r = blockIdx.x;
  const int tid = threadIdx.x;
  const float* c = content + (size_t)r * Dn;
  for (int i = tid; i < Dn; i += 256) row[i] = c[i];
  __syncthreads();
  const int o = tid >> 3, sub = tid & 7;
  const float* w = (o < Hn) ? (b_w + (size_t)o * Dn) : (ts_w + (size_t)(o - Hn) * Dn);
  float s = 0.f;
  for (int d = sub; d < Dn; d += 8) s += row[d] * w[d];
  part[o][sub] = s;
  __syncthreads();
  if (tid < 32) {
    float t = 0.f;
#pragma unroll
    for (int i = 0; i < 8; i++) t += part[tid][i];
    if (tid < Hn) bl[(size_t)r * Hn + tid] = t + b_b[tid];
    else          gl[(size_t)r * Hn + (tid - Hn)] = t + ts_b[tid - Hn] + hd[tid - Hn];
  }
}

// ---------------------------------------------------------------------------
// Kernel 3: per-(b,h) chunked delta-rule write. 512 blocks x 256 threads.
// k/v tiles staged by the Tensor Data Mover (strided 2-D tile + LDS padding).
// ---------------------------------------------------------------------------
__global__ __launch_bounds__(256) void chunk_kernel(const float* __restrict__ W_old,
                                                    const float* __restrict__ pk,
                                                    const float* __restrict__ pv,
                                                    const float* __restrict__ bl,
                                                    const float* __restrict__ gl,
                                                    float* __restrict__ Wnew) {
  __shared__ float Ks[Tn][66];        // normalized k, row-major [t][d] (pad 2)
  __shared__ float Vs[Tn][66];        // raw v -> Vrhs -> Vp in place
  __shared__ float Wt[DKn][66];       // W_old transposed: Wt[d][v]
  __shared__ float Ablk[4][16][18];
  __shared__ float Adiag[16][18];
  __shared__ float betaS[Tn];
  __shared__ float cumS[Tn];
  __shared__ float wS[Tn];

  const int bh = blockIdx.x;
  const int b = bh >> 4, h = bh & 15;
  const int tid = threadIdx.x;
  const int lane = tid & 31;
  const int wave = tid >> 5;
  const int ml = lane & 15;
  const int hi = lane >> 4;
  const int kb = hi * 2;

#ifdef HAS_TDM
  // ---- issue TDM tile loads early; overlap with beta/gamma/scan ----
  // tile: 256 rows x 64 floats, row stride 1024 floats, LDS pad 2 DW / 64 DW
  if (tid == 0) {
    const unsigned long long gk =
        (unsigned long long)(uintptr_t)(pk + (size_t)(b * Tn) * PROJ_N + h * DKn);
    const unsigned long long gv =
        (unsigned long long)(uintptr_t)(pv + (size_t)(b * Tn) * PROJ_N + h * DKn);
    tdm_load_2d(lds_offset_of(&Ks[0][0]), gk, DKn, Tn, PROJ_N, 5, 1, 1);
    tdm_load_2d(lds_offset_of(&Vs[0][0]), gv, DKn, Tn, PROJ_N, 5, 1, 1);
  }
#endif

  // ---- phase 1: beta, gamma, inclusive scan of log(gamma) ----
  {
    const float blv = bl[(size_t)(b * Tn + tid) * Hn + h];
    const float glv = gl[(size_t)(b * Tn + tid) * Hn + h];
    betaS[tid] = 1.f / (1.f + __expf(-blv));
    const float gma = 1.f / (1.f + __expf(-glv));
    cumS[tid] = __logf(fmaxf(gma, 1e-8f));
  }
  __syncthreads();
  for (int off = 1; off < Tn; off <<= 1) {
    float v = cumS[tid];
    if (tid >= off) v += cumS[tid - off];
    __syncthreads();
    cumS[tid] = v;
    __syncthreads();
  }

#ifdef HAS_TDM
  if (tid == 0) WAIT_TENSOR0();
  __syncthreads();
  // ---- phase 2: L2-normalize k rows in LDS ----
  {
    float s = 0.f;
#pragma unroll
    for (int d = 0; d < DKn; d++) { const float q = Ks[tid][d]; s += q * q; }
    const float inv = 1.f / fmaxf(sqrtf(s), 1e-12f);
#pragma unroll
    for (int d = 0; d < DKn; d++) Ks[tid][d] *= inv;
  }
#else
  // ---- fallback: manual staging of k (normalized) and raw v ----
  {
    const float* src = pk + (size_t)(b * Tn + tid) * PROJ_N + h * DKn;
    const float* vsrc = pv + (size_t)(b * Tn + tid) * PROJ_N + h * DKn;
    float s = 0.f;
#pragma unroll
    for (int d = 0; d < DKn; d += 4) {
      const float4 q = *(const float4*)(src + d);
      Ks[tid][d + 0] = q.x; Ks[tid][d + 1] = q.y;
      Ks[tid][d + 2] = q.z; Ks[tid][d + 3] = q.w;
      s += q.x * q.x + q.y * q.y + q.z * q.z + q.w * q.w;
      const float4 qv = *(const float4*)(vsrc + d);
      Vs[tid][d + 0] = qv.x; Vs[tid][d + 1] = qv.y;
      Vs[tid][d + 2] = qv.z; Vs[tid][d + 3] = qv.w;
    }
    const float inv = 1.f / fmaxf(sqrtf(s), 1e-12f);
#pragma unroll
    for (int d = 0; d < DKn; d++) Ks[tid][d] *= inv;
  }
#endif

  // ---- phase 3: stage W_old transposed ----
  const float* W0 = W_old + (size_t)bh * (DKn * DKn);
  for (int i = tid; i < DKn * DKn; i += 256) {
    const int vv = i >> 6, dd = i & 63;
    Wt[dd][vv] = W0[i];
  }
  __syncthreads();

  // ---- phase 4: Vrhs = v - G_{i-1} * (W_old k)  via WMMA ----
#pragma unroll
  for (int mt2 = 0; mt2 < 2; mt2++) {
    const int m0 = (wave * 2 + mt2) * 16;
    v8f acc[4];
#pragma unroll
    for (int n = 0; n < 4; n++) acc[n] = vz8();
    for (int kk = 0; kk < DKn; kk += 4) {
      const v2f a = *(const v2f*)&Ks[m0 + ml][kk + kb];
#pragma unroll
      for (int n = 0; n < 4; n++) {
        v2f bf = { Wt[kk + kb][n * 16 + ml], Wt[kk + kb + 1][n * 16 + ml] };
        acc[n] = wmma4(a, bf, acc[n]);
      }
    }
#pragma unroll
    for (int n = 0; n < 4; n++)
#pragma unroll
      for (int r = 0; r < 8; r++) {
        const int row = m0 + r + hi * 8;
        const int col = n * 16 + ml;
        const float vv = Vs[row][col];
        const float gim1 = (row == 0) ? 1.f : __expf(cumS[row - 1]);
        Vs[row][col] = vv - gim1 * acc[n][r];
      }
  }
  __syncthreads();

  // ---- phase 5: blocked unit-lower forward substitution ----
  for (int I = 0; I < 16; I++) {
    const int i0 = I * 16;
    if (I > 0 && wave < 4) {
      const int n0 = wave * 16;
      v8f S = vz8();
      for (int J = 0; J < I; J++) {
        const int j0 = J * 16;
        v8f kkacc = vz8();
#pragma unroll
        for (int kk = 0; kk < DKn; kk += 4) {
          const v2f a  = *(const v2f*)&Ks[i0 + ml][kk + kb];
          const v2f bf = *(const v2f*)&Ks[j0 + ml][kk + kb];
          kkacc = wmma4(a, bf, kkacc);
        }
#pragma unroll
        for (int r = 0; r < 8; r++) {
          const int ri = i0 + r + hi * 8;
          const int c  = j0 + ml;
          Ablk[wave][r + hi * 8][ml] =
              betaS[c] * __expf(cumS[ri - 1] - cumS[c]) * kkacc[r];
        }
#pragma unroll
        for (int kk = 0; kk < 16; kk += 4) {
          const v2f a2 = *(const v2f*)&Ablk[wave][ml][kk + kb];
          v2f bf2 = { Vs[j0 + kk + kb][n0 + ml], Vs[j0 + kk + kb + 1][n0 + ml] };
          S = wmma4(a2, bf2, S);
        }
      }
#pragma unroll
      for (int r = 0; r < 8; r++) Vs[i0 + r + hi * 8][n0 + ml] -= S[r];
    }
    if (wave == 4) {
      v8f kkacc = vz8();
#pragma unroll
      for (int kk = 0; kk < DKn; kk += 4) {
        const v2f a  = *(const v2f*)&Ks[i0 + ml][kk + kb];
        const v2f bf = *(const v2f*)&Ks[i0 + ml][kk + kb];
        kkacc = wmma4(a, bf, kkacc);
      }
#pragma unroll
      for (int r = 0; r < 8; r++) {
        const int rr = r + hi * 8;
        float aval = 0.f;
        if (ml < rr) {
          const int ri = i0 + rr, c = i0 + ml;
          aval = betaS[c] * __expf(cumS[ri - 1] - cumS[c]) * kkacc[r];
        }
        Adiag[rr][ml] = aval;
      }
    }
    __syncthreads();
    if (tid < DKn) {
      float vloc[16];
#pragma unroll
      for (int r = 0; r < 16; r++) vloc[r] = Vs[i0 + r][tid];
#pragma unroll
      for (int r = 1; r < 16; r++) {
        float s = vloc[r];
        for (int c = 0; c < r; c++) s -= Adiag[r][c] * vloc[c];
        vloc[r] = s;
      }
#pragma unroll
      for (int r = 0; r < 16; r++) Vs[i0 + r][tid] = vloc[r];
    }
    __syncthreads();
  }

  // ---- phase 6: W_new = G_T * W_old + sum_t (beta*decay*Vp)[t,v] k[t,d] ----
  wS[tid] = betaS[tid] * __expf(cumS[Tn - 1] - cumS[tid]);
  __syncthreads();
  {
    const float gT = __expf(cumS[Tn - 1]);
#pragma unroll
    for (int i2 = 0; i2 < 2; i2++) {
      const int tile = wave * 2 + i2;
      const int m0 = (tile >> 2) * 16;
      const int n0 = (tile & 3) * 16;
      v8f acc = vz8();
      for (int kk = 0; kk < Tn; kk += 4) {
        const int kx = kk + kb;
        v2f a  = { wS[kx] * Vs[kx][m0 + ml], wS[kx + 1] * Vs[kx + 1][m0 + ml] };
        v2f bf = { Ks[kx][n0 + ml],          Ks[kx + 1][n0 + ml] };
        acc = wmma4(a, bf, acc);
      }
#pragma unroll
      for (int r = 0; r < 8; r++) {
        const int vv = m0 + r + hi * 8, dd = n0 + ml;
        Wnew[(size_t)bh * (DKn * DKn) + vv * DKn + dd] =
            gT * W0[vv * DKn + dd] + acc[r];
      }
    }
  }
}

// ---------------------------------------------------------------------------
extern "C" void kernel_launch(void* const* d_in, const int* in_sizes, int n_in,
                              void* d_out, int out_size, void* d_ws, size_t ws_size,
                              hipStream_t stream) {
  const float* W_old   = (const float*)d_in[0];
  const float* content = (const float*)d_in[1];
  const float* k_w     = (const float*)d_in[2];
  const float* v_w     = (const float*)d_in[3];
  const float* b_w     = (const float*)d_in[4];
  const float* b_b     = (const float*)d_in[5];
  const float* ts_w    = (const float*)d_in[6];
  const float* ts_b    = (const float*)d_in[7];
  const float* hd      = (const float*)d_in[8];
  float* out = (float*)d_out;

  float* pk  = (float*)d_ws;
  float* pv  = pk + (size_t)(Bn * Tn) * PROJ_N;
  float* blg = pv + (size_t)(Bn * Tn) * PROJ_N;
  float* glg = blg + (size_t)(Bn * Tn) * Hn;

  dim3 g((Bn * Tn) / 32, PROJ_N / 256);
  proj_gemm<<<g, 256, 0, stream>>>(content, k_w, pk);
  proj_gemm<<<g, 256, 0, stream>>>(content, v_w, pv);
  proj_small<<<Bn * Tn, 256, 0, stream>>>(content, b_w, b_b, ts_w, ts_b, hd, blg, glg);
  chunk_kernel<<<Bn * Hn, 256, 0, stream>>>(W_old, pk, pv, blg, glg, out);
}